// DGRecLayer_80410377716439
// MI455X (gfx1250) — compile-verified
//
#include <hip/hip_runtime.h>

typedef __attribute__((ext_vector_type(2))) float v2f;
typedef __attribute__((ext_vector_type(8))) float v8f;

#define NUM_SRC 16384
#define NUM_DST 8192
#define N_NBR   50
#define D_FEAT  64
#define K_SEL   10
#define ROWP    52   // padded sims row stride / padded K extent (4-aligned)

// ---------------- Kernel 1: zero the degree histogram ----------------
__global__ void dgrec_zero_deg(unsigned* __restrict__ deg) {
    int i = blockIdx.x * blockDim.x + threadIdx.x;
    if (i < NUM_SRC) deg[i] = 0u;
}

// ---------------- Kernel 2: degree histogram over flat neighbor_idx ----------------
__global__ void dgrec_histogram(const int* __restrict__ nbr, unsigned* __restrict__ deg, int n) {
    int i = blockIdx.x * blockDim.x + threadIdx.x;
    if (i < n) atomicAdd(&deg[nbr[i]], 1u);
}

// ---------------- Kernel 3: fused gather + greedy + WMMA feature contraction ----------------
// One block (128 threads = 4 wave32) per destination row.
__global__ __launch_bounds__(128) void dgrec_main(
    const float* __restrict__ h_src,
    const float* __restrict__ sims_table,
    const int*   __restrict__ nbr,
    const unsigned* __restrict__ deg,
    float* __restrict__ out)
{
    __shared__ float s_sims[N_NBR * ROWP];
    __shared__ float s_cache[ROWP];
    __shared__ float s_gain[ROWP];
    __shared__ float s_v[ROWP];
    __shared__ int   s_idx[ROWP];   // padded: [50],[51] = 0 (safe row, zero weight)
    __shared__ int   s_off[ROWP];   // idx * D_FEAT, premultiplied row offsets
    __shared__ int   s_w[ROWP];
    __shared__ int   s_sel;

    const int b = blockIdx.x;
    const int t = threadIdx.x;

    // ---- Phase 1: load neighbor indices (padded), init state ----
    if (t < ROWP) {
        int idx = (t < N_NBR) ? nbr[b * N_NBR + t] : 0;
        s_idx[t]   = idx;
        s_off[t]   = idx * D_FEAT;
        s_cache[t] = 0.0f;
        s_w[t]     = 0;
        s_v[t]     = 0.0f;
    }
    __syncthreads();

    // ---- Phase 2: gather 50x50 sims tile into LDS ----
    for (int e = t; e < N_NBR * N_NBR; e += 128) {
        int i = e / N_NBR;
        int j = e - i * N_NBR;
        size_t off = (size_t)s_idx[i] * (size_t)NUM_SRC + (size_t)s_idx[j];
        s_sims[i * ROWP + j] = sims_table[off];
    }
    __syncthreads();

    // ---- Phase 3: greedy max-coverage scan (exact first-argmax semantics) ----
    for (int k = 0; k < K_SEL; ++k) {
        if (t < N_NBR) {
            const float* row = &s_sims[t * ROWP];
            float g = 0.0f;
            for (int j = 0; j < N_NBR; ++j) {
                float c = s_cache[j];
                g += fmaxf(row[j], c) - c;
            }
            s_gain[t] = g;
        }
        __syncthreads();
        if (t == 0) {
            float best = s_gain[0];
            int sel = 0;
            for (int i = 1; i < N_NBR; ++i) {
                if (s_gain[i] > best) { best = s_gain[i]; sel = i; } // strict > => first max
            }
            s_sel = sel;
            s_w[sel] += 1;
        }
        __syncthreads();
        if (t < N_NBR) {
            s_cache[t] = fmaxf(s_sims[s_sel * ROWP + t], s_cache[t]);
        }
        __syncthreads();
    }

    // ---- Phase 4: weight vector v[j] = count_j * norm_out(idx_j) ----
    if (t < N_NBR) {
        int w = s_w[t];
        if (w > 0) {
            float d = (float)deg[s_idx[t]];
            s_v[t] = (float)w * rsqrtf(fmaxf(d, 1.0f));
        }
    }
    __syncthreads();

    // ---- Phase 5: out^T (64x1) = feat^T (64x50) x v (50x1) via V_WMMA_F32_16X16X4_F32 ----
    // 4 waves, each owns a 16-row M-tile of the 64 feature dims; 13 K-steps of 4 (50 -> 52 padded,
    // pad rows carry zero weight so A may load the safe row unconditionally).
    // A (16x4 f32, 2 VGPRs/lane): lanes 0-15 hold K=0(v0),K=1(v1); lanes 16-31 hold K=2(v0),K=3(v1).
    // B (4x16 f32, 2 VGPRs/lane, mirrored striping): only column N=0 carries v (cndmask select).
    // D (16x16 f32, 8 VGPRs): column N=0 lives in lanes 0 (M=r) and 16 (M=8+r).
    const int wave   = t >> 5;
    const int lane   = t & 31;
    const int m      = lane & 15;        // N / M-within-half index
    const int khalf  = lane >> 4;        // 0: K pair (0,1); 1: K pair (2,3)
    const int m_base = wave * 16;
    const bool col0  = (m == 0);

    const float* __restrict__ hp = h_src + m_base + m;   // per-lane feature column base

    v8f c = {};
    const float inv_sqrt_n = 0.14142135623730950488f; // 50^-0.5

    #pragma unroll
    for (int kk = 0; kk < 13; ++kk) {
        const int kA0 = kk * 4 + khalf * 2;
        const int kA1 = kA0 + 1;

        v2f a;
        a.x = hp[s_off[kA0]];            // unconditional: pad rows are idx 0, weight 0
        a.y = hp[s_off[kA1]];

        const float v0 = s_v[kA0];       // unconditional LDS read -> register
        const float v1 = s_v[kA1];
        v2f bm;
        bm.x = col0 ? v0 : 0.0f;         // v_cndmask, no exec branching
        bm.y = col0 ? v1 : 0.0f;

        c = __builtin_amdgcn_wmma_f32_16x16x4_f32(
                /*neg_a=*/false, a, /*neg_b=*/false, bm,
                /*c_mod=*/(short)0, c, /*reuse_a=*/false, /*reuse_b=*/false);
    }

    // Column 0 of D: lane 0 -> M=r, lane 16 -> M=8+r (r = VGPR index 0..7)
    if (col0) {
        const int mhi = khalf * 8;
        float* orow = out + (size_t)b * D_FEAT + m_base + mhi;
        #pragma unroll
        for (int r = 0; r < 8; ++r) {
            orow[r] = c[r] * inv_sqrt_n;
        }
    }
}

extern "C" void kernel_launch(void* const* d_in, const int* in_sizes, int n_in,
                              void* d_out, int out_size, void* d_ws, size_t ws_size,
                              hipStream_t stream) {
    const float* h_src      = (const float*)d_in[0];
    const float* sims_table = (const float*)d_in[1];
    const int*   nbr        = (const int*)d_in[2];
    float* out = (float*)d_out;
    unsigned* deg = (unsigned*)d_ws;   // 16384 * 4 B of scratch

    const int n_flat = NUM_DST * N_NBR;

    dgrec_zero_deg<<<(NUM_SRC + 255) / 256, 256, 0, stream>>>(deg);
    dgrec_histogram<<<(n_flat + 255) / 256, 256, 0, stream>>>(nbr, deg, n_flat);
    dgrec_main<<<NUM_DST, 128, 0, stream>>>(h_src, sims_table, nbr, deg, out);
}